// MHSA3D_15745350107203
// MI455X (gfx1250) — compile-verified
//
#include <hip/hip_runtime.h>

typedef __attribute__((ext_vector_type(2))) float v2f;
typedef __attribute__((ext_vector_type(8))) float v8f;

#define NHEADS 8
#define DH     16
#define NTOK   4096   // 16*16*16 tokens
#define CIN    128
#define OQKV   384    // 3 * 128

__device__ __forceinline__ v8f wmma_f32_16x16x4(v2f a, v2f b, v8f c) {
  // (neg_a, A, neg_b, B, c_mod, C, reuse_a, reuse_b)
  return __builtin_amdgcn_wmma_f32_16x16x4_f32(false, a, false, b, (short)0, c,
                                               false, false);
}

// ---------------------------------------------------------------------------
// Kernel 1: QKV projection (1x1x1 conv == GEMM), WMMA f32 16x16x4.
//   qkv[o][n] = sum_c w[c][o] * x[c][n] + b[o]
// Epilogue re-layouts for the attention kernel's fragment loads:
//   Q: *dh^-0.5, stored d-pair interleaved  [h][d/2][n][2]   (b64 frag loads)
//   K: + decomposed positional emb, same d-pair interleave
//   V: stored n-pair interleaved            [h][n/2][d][2]   (b64 frag loads)
// One 16(o) x 16(n) tile per wave. 24 * 256 = 6144 tiles, 8 waves/block.
// ---------------------------------------------------------------------------
__global__ __launch_bounds__(256)
void qkv_proj_kernel(const float* __restrict__ x,  const float* __restrict__ w,
                     const float* __restrict__ bq, const float* __restrict__ ed,
                     const float* __restrict__ eh, const float* __restrict__ ew,
                     float* __restrict__ qs2, float* __restrict__ ks2,
                     float* __restrict__ vs2)
{
  const int lane = threadIdx.x & 31;
  const int half = lane >> 4;        // 0 or 1 (K sub-chunk select)
  const int m    = lane & 15;        // row/col within tile
  const int wid  = blockIdx.x * 8 + (threadIdx.x >> 5);
  const int o0   = (wid >> 8) * 16;  // 24 o-tiles
  const int n0   = (wid & 255) * 16;

  v8f acc = {};
  #pragma unroll 4
  for (int c0 = 0; c0 < CIN; c0 += 4) {
    const int ca = c0 + 2 * half;
    v2f a, b;
    a.x = w[(ca    ) * OQKV + o0 + m];   // A[m][k] = w[c][o], coalesced in o
    a.y = w[(ca + 1) * OQKV + o0 + m];
    b.x = x[(ca    ) * NTOK + n0 + m];   // B[k][n] = x[c][n], coalesced in n
    b.y = x[(ca + 1) * NTOK + n0 + m];
    acc = wmma_f32_16x16x4(a, b, acc);
  }

  const int n  = n0 + m;
  const int dz = n >> 8, hy = (n >> 4) & 15, wx = n & 15;
  // C/D layout: row o = o0 + r + 8*half. r-pairs (r, r+1) are adjacent o's,
  // which land in adjacent floats of the d-pair-interleaved Q/K layouts.
  #pragma unroll
  for (int r = 0; r < 8; r += 2) {
    const int o = o0 + r + 8 * half;               // even
    float v0 = acc[r]     + bq[o];
    float v1 = acc[r + 1] + bq[o + 1];
    if (o < 128) {                                 // Q
      v2f qv = {v0 * 0.25f, v1 * 0.25f};           // fold dh^-0.5
      ((v2f*)qs2)[(o >> 1) * NTOK + n] = qv;
    } else if (o < 256) {                          // K + positional bias
      const int oo = o - 128;
      const int d0 = oo & 15;                      // even; d1 = d0+1
      v0 += ed[d0 * 16 + dz] + eh[d0 * 16 + hy] + ew[d0 * 16 + wx];
      v1 += ed[(d0 + 1) * 16 + dz] + eh[(d0 + 1) * 16 + hy] + ew[(d0 + 1) * 16 + wx];
      v2f kv = {v0, v1};
      ((v2f*)ks2)[(oo >> 1) * NTOK + n] = kv;
    } else {                                       // V: [h][n/2][d][2]
      const int oo = o - 256;
      const int h  = oo >> 4, d = oo & 15;
      const int base = ((h * (NTOK / 2) + (n >> 1)) * DH + d) * 2 + (n & 1);
      vs2[base]          = v0;                     // d
      vs2[base + 2]      = v1;                     // d+1 -> +2 floats
    }
  }
}

// ---------------------------------------------------------------------------
// Kernel 2: flash attention, fp32 WMMA. One 16-query tile per wave.
// 8 heads * 256 q-tiles = 2048 wave-tasks -> 256 blocks * 8 waves.
// 64 columns per round: 16 WMMA (S = Q^T K'), one online-softmax round
// (elementwise-combine 4 tiles, then one 16-lane shuffle reduction),
// P transposed C/D->A layout via per-wave LDS (b64 reads), 16 WMMA (O += P V^T).
// Next round's K'/V lines are prefetched (global_prefetch_b8) to warm the
// WGP$ one iteration ahead; all 8 waves of a block share the same head.
// ---------------------------------------------------------------------------
__global__ __launch_bounds__(256)
void attn_kernel(const float* __restrict__ qs2, const float* __restrict__ ks2,
                 const float* __restrict__ vs2, float* __restrict__ out)
{
  __shared__ float smem[8 * 16 * 64];    // 4KB per wave: 16x64 P transpose
  const int lane = threadIdx.x & 31;
  const int half = lane >> 4;
  const int m    = lane & 15;
  const int wave = threadIdx.x >> 5;
  const int h    = blockIdx.x >> 5;
  const int i0   = ((blockIdx.x & 31) * 8 + wave) * 16;
  float* sw = smem + wave * (16 * 64);

  const v2f* qp = (const v2f*)qs2;       // [h][d/2][n]{2}
  const v2f* kp = (const v2f*)ks2;
  const v2f* vp = (const v2f*)vs2;       // [h][n/2][d]{2}

  v2f qa[4];                             // Q tile as A-frags, held in regs
  #pragma unroll
  for (int s = 0; s < 4; ++s)
    qa[s] = qp[(h * 8 + s * 2 + half) * NTOK + i0 + m];

  v8f acc = {};
  float rm[8], rs[8];
  #pragma unroll
  for (int r = 0; r < 8; ++r) { rm[r] = -3.0e38f; rs[r] = 0.0f; }

  for (int j0 = 0; j0 < NTOK; j0 += 64) {
    // Prefetch next iteration's K'/V streams into the WGP cache.
    // Speculative; OOB at the tail is silently dropped / stays in-allocation.
    #pragma unroll
    for (int s = 0; s < 4; ++s)
      __builtin_prefetch(&kp[(h * 8 + s * 2 + half) * NTOK + j0 + 64 + s * 16 + m], 0, 1);
    #pragma unroll
    for (int u = 0; u < 16; u += 4)
      __builtin_prefetch(&vp[(h * (NTOK / 2) + ((j0 + 64) >> 1) + u * 2 + half) * DH + m], 0, 1);

    // S[t] = Q^T K' for 4 adjacent 16-col tiles (K' has emb, Q has scale)
    v8f S[4];
    #pragma unroll
    for (int t = 0; t < 4; ++t) {
      S[t] = (v8f){};
      #pragma unroll
      for (int s = 0; s < 4; ++s) {
        v2f kb = kp[(h * 8 + s * 2 + half) * NTOK + j0 + t * 16 + m];
        S[t] = wmma_f32_16x16x4(qa[s], kb, S[t]);
      }
    }
    // Online softmax across all 64 columns at once.
    // Row i = r + 8*half lives in vgpr r across a 16-lane half.
    #pragma unroll
    for (int r = 0; r < 8; ++r) {
      float mx = fmaxf(fmaxf(S[0][r], S[1][r]), fmaxf(S[2][r], S[3][r]));
      mx = fmaxf(mx, __shfl_xor(mx, 1, 32));
      mx = fmaxf(mx, __shfl_xor(mx, 2, 32));
      mx = fmaxf(mx, __shfl_xor(mx, 4, 32));
      mx = fmaxf(mx, __shfl_xor(mx, 8, 32));
      const float nm = fmaxf(rm[r], mx);
      const float sc = __expf(rm[r] - nm);
      float sm = 0.0f;
      #pragma unroll
      for (int t = 0; t < 4; ++t) {
        const float p = __expf(S[t][r] - nm);      // TRANS, co-executes
        S[t][r] = p;
        sm += p;
      }
      sm += __shfl_xor(sm, 1, 32);
      sm += __shfl_xor(sm, 2, 32);
      sm += __shfl_xor(sm, 4, 32);
      sm += __shfl_xor(sm, 8, 32);
      rs[r]  = rs[r] * sc + sm;
      rm[r]  = nm;
      acc[r] *= sc;                      // O rows share the same row layout
    }
    // Transpose P: C/D layout -> A layout via per-wave LDS scratch.
    // LDS is in-order within a wave; the wait orders cross-lane consumption.
    #pragma unroll
    for (int t = 0; t < 4; ++t)
      #pragma unroll
      for (int r = 0; r < 8; ++r)
        sw[(r + 8 * half) * 64 + t * 16 + m] = S[t][r];
    asm volatile("s_wait_dscnt 0" ::: "memory");
    // O += P @ V^T : 16 K-steps over 64 columns; all frags are b64 loads
    const v2f* swp = (const v2f*)sw;
    #pragma unroll
    for (int u = 0; u < 16; ++u) {
      v2f pa = swp[m * 32 + u * 2 + half];                       // ds b64
      v2f vb = vp[(h * (NTOK / 2) + (j0 >> 1) + u * 2 + half) * DH + m];
      acc = wmma_f32_16x16x4(pa, vb, acc);
    }
  }
  // Normalize, then transpose O through LDS for coalesced output stores.
  #pragma unroll
  for (int r = 0; r < 8; ++r)
    sw[(r + 8 * half) * 16 + m] = acc[r] / rs[r];  // [i_local][d]
  asm volatile("s_wait_dscnt 0" ::: "memory");
  #pragma unroll
  for (int r = 0; r < 8; ++r) {
    const int d = r + 8 * half;
    out[(h * DH + d) * NTOK + i0 + m] = sw[m * 16 + d];  // 64B/half-wave
  }
}

extern "C" void kernel_launch(void* const* d_in, const int* in_sizes, int n_in,
                              void* d_out, int out_size, void* d_ws, size_t ws_size,
                              hipStream_t stream) {
  const float* x  = (const float*)d_in[0];
  const float* w  = (const float*)d_in[1];
  const float* b  = (const float*)d_in[2];
  const float* ed = (const float*)d_in[3];
  const float* eh = (const float*)d_in[4];
  const float* ew = (const float*)d_in[5];
  float* out = (float*)d_out;

  float* qs2 = (float*)d_ws;             // [8][8][4096][2] scaled Q     (2 MB)
  float* ks2 = qs2 + CIN * NTOK;         // [8][8][4096][2] K + emb      (2 MB)
  float* vs2 = ks2 + CIN * NTOK;         // [8][2048][16][2] V n-paired  (2 MB)

  qkv_proj_kernel<<<768, 256, 0, stream>>>(x, w, b, ed, eh, ew, qs2, ks2, vs2);
  attn_kernel<<<256, 256, 0, stream>>>(qs2, ks2, vs2, out);
}